// UTt_69166153335437
// MI455X (gfx1250) — compile-verified
//
#include <hip/hip_runtime.h>
#include <math.h>

// ---------------------------------------------------------------------------
// MoE 2-layer MLP (DIM=32) with per-token expert routing, for gfx1250.
// Memory-bound (~33 MB traffic, ~0.54 GFLOP) -> f32 WMMA (16x16x4) is the
// right matrix path: full accuracy, compute is nowhere near the bottleneck.
// Pipeline: histogram -> 16-aligned scan -> scatter token ids -> grouped GEMM
// where each wave32 owns one 16-token single-expert tile and issues 32
// v_wmma_f32_16x16x4_f32 ops (2 layers x 2 col-blocks x 8 K-chunks).
// ---------------------------------------------------------------------------

typedef float v2f __attribute__((ext_vector_type(2)));
typedef float v8f __attribute__((ext_vector_type(8)));

#define DIM 32
#define WAVES_PER_BLOCK 8
#define LDS_STRIDE 36   // floats; 36*4=144B keeps float2/float4 rows aligned

// ws layout (ints): [0..31] counts, [32..63] cursors, [64..127] poffs(E+1),
// [128 ..) bucket (B entries)
__global__ void init_ws_kernel(int* ws) {
    if (threadIdx.x < 128) ws[threadIdx.x] = 0;
}

__global__ void count_kernel(const int* __restrict__ pos,
                             int* __restrict__ counts, int B) {
    int t = blockIdx.x * blockDim.x + threadIdx.x;
    if (t < B) atomicAdd(&counts[pos[t]], 1);
}

// 16-align each expert segment so every 16-token tile is single-expert.
__global__ void scan_kernel(const int* __restrict__ counts,
                            int* __restrict__ poffs, int E) {
    if (threadIdx.x == 0) {
        int acc = 0;
        poffs[0] = 0;
        for (int e = 0; e < E; ++e) {
            acc += (counts[e] + 15) & ~15;
            poffs[e + 1] = acc;
        }
    }
}

__global__ void scatter_kernel(const int* __restrict__ pos,
                               const int* __restrict__ poffs,
                               int* __restrict__ cursors,
                               int* __restrict__ bucket, int B) {
    int t = blockIdx.x * blockDim.x + threadIdx.x;
    if (t < B) {
        int e = pos[t];
        int idx = atomicAdd(&cursors[e], 1);
        bucket[poffs[e] + idx] = t;
    }
}

__global__ __launch_bounds__(WAVES_PER_BLOCK * 32)
void moe_wmma_kernel(const float* __restrict__ x,
                     const float* __restrict__ W1, const float* __restrict__ b1,
                     const float* __restrict__ W0, const float* __restrict__ b0,
                     const int* __restrict__ bucket,
                     const int* __restrict__ counts,
                     const int* __restrict__ poffs,
                     float* __restrict__ out, int E) {
    __shared__ float lds_h[WAVES_PER_BLOCK * 16 * LDS_STRIDE];
    __shared__ float lds_o[WAVES_PER_BLOCK * 16 * LDS_STRIDE];

    const int lane = threadIdx.x & 31;
    const int wave = threadIdx.x >> 5;
    const int gt    = blockIdx.x * WAVES_PER_BLOCK + wave;  // global tile id
    const int tbase = gt << 4;
    const int total = poffs[E];
    if (tbase >= total) return;          // wave-uniform: EXEC stays all-1s

    // Find owning expert (E<=32, uniform scalar-ish loop).
    int e = 0;
    while (poffs[e + 1] <= tbase) ++e;
    const int segStart = poffs[e];
    const int cnt      = counts[e];
    const int li       = tbase - segStart;

    const int slot = lane & 15;          // token slot / row m / col n role
    const int hf   = lane >> 4;          // lane half -> K sub-pair in A/B frags
    const bool valid = (li + slot) < cnt;
    const int tok = bucket[segStart + (valid ? (li + slot) : 0)];

    // A-matrix source: lane m reads float2 of its token row; lanes 16-31 get K+2.
    const v2f* xr  = (const v2f*)(x + (size_t)tok * DIM);
    // B-matrix source: B[k,n] = W[n,k]; lane n reads float2 from row n of W.
    const v2f* w1a = (const v2f*)(W1 + ((size_t)e * DIM + slot) * DIM);
    const v2f* w1b = (const v2f*)(W1 + ((size_t)e * DIM + 16 + slot) * DIM);

    // ---- Layer 1: H(16x32) = X(16x32) * W1^T ----
    v8f acc0 = {}; v8f acc1 = {};
#pragma unroll
    for (int kc = 0; kc < 8; ++kc) {
        v2f a  = xr[2 * kc + hf];
        acc0 = __builtin_amdgcn_wmma_f32_16x16x4_f32(false, a, false,
                 w1a[2 * kc + hf], (short)0, acc0, false, false);
        acc1 = __builtin_amdgcn_wmma_f32_16x16x4_f32(false, a, false,
                 w1b[2 * kc + hf], (short)0, acc1, false, false);
    }
    const float bias0 = b1[e * DIM + slot];
    const float bias1 = b1[e * DIM + 16 + slot];

    // D-layout -> row-major 16x32 tile in LDS (per-wave region, in-order DS).
    float* H = lds_h + wave * (16 * LDS_STRIDE);
    const int mbase = hf * 8;            // lanes 16-31 hold rows M+8
#pragma unroll
    for (int v = 0; v < 8; ++v) {
        H[(mbase + v) * LDS_STRIDE + slot]      = acc0[v] + bias0;
        H[(mbase + v) * LDS_STRIDE + 16 + slot] = acc1[v] + bias1;
    }

    // ---- Layer 2: OUT(16x32) = H(16x32) * W0^T ----
    const float* Hr = H + slot * LDS_STRIDE;        // row m = slot
    const v2f* w0a = (const v2f*)(W0 + ((size_t)e * DIM + slot) * DIM);
    const v2f* w0b = (const v2f*)(W0 + ((size_t)e * DIM + 16 + slot) * DIM);
    v8f o0 = {}; v8f o1 = {};
#pragma unroll
    for (int kc = 0; kc < 8; ++kc) {
        v2f a = *(const v2f*)(Hr + 4 * kc + 2 * hf);
        o0 = __builtin_amdgcn_wmma_f32_16x16x4_f32(false, a, false,
               w0a[2 * kc + hf], (short)0, o0, false, false);
        o1 = __builtin_amdgcn_wmma_f32_16x16x4_f32(false, a, false,
               w0b[2 * kc + hf], (short)0, o1, false, false);
    }
    const float c0 = b0[e * DIM + slot];
    const float c1 = b0[e * DIM + 16 + slot];

    float* O = lds_o + wave * (16 * LDS_STRIDE);
#pragma unroll
    for (int v = 0; v < 8; ++v) {
        O[(mbase + v) * LDS_STRIDE + slot]      = o0[v] + c0;
        O[(mbase + v) * LDS_STRIDE + 16 + slot] = o1[v] + c1;
    }

    // ---- L2 normalize + scatter store: lane (slot,hf) owns 16 dims of token.
    const float* Or = O + slot * LDS_STRIDE + hf * 16;  // 16B-aligned
    float r[16];
    float ss = 0.f;
#pragma unroll
    for (int i = 0; i < 16; ++i) { r[i] = Or[i]; ss += r[i] * r[i]; }
    float tot   = ss + __shfl_xor(ss, 16, 32);          // combine dim halves
    float scale = 1.0f / fmaxf(sqrtf(tot), 1e-12f);

    if (valid) {
        float4* op = (float4*)(out + (size_t)tok * DIM + hf * 16);
#pragma unroll
        for (int i = 0; i < 4; ++i) {
            float4 vv;
            vv.x = r[4 * i + 0] * scale;
            vv.y = r[4 * i + 1] * scale;
            vv.z = r[4 * i + 2] * scale;
            vv.w = r[4 * i + 3] * scale;
            op[i] = vv;
        }
    }
}

extern "C" void kernel_launch(void* const* d_in, const int* in_sizes, int n_in,
                              void* d_out, int out_size, void* d_ws, size_t ws_size,
                              hipStream_t stream) {
    const float* x  = (const float*)d_in[0];
    const float* W1 = (const float*)d_in[1];
    const float* b1 = (const float*)d_in[2];
    const float* W0 = (const float*)d_in[3];
    const float* b0 = (const float*)d_in[4];
    const int*   pos = (const int*)d_in[5];

    const int B = in_sizes[0] / DIM;     // tokens
    const int E = in_sizes[2] / DIM;     // experts (b1 is E*DIM)

    int* ws      = (int*)d_ws;
    int* counts  = ws;
    int* cursors = ws + 32;
    int* poffs   = ws + 64;
    int* bucket  = ws + 128;
    (void)ws_size; (void)n_in; (void)out_size;

    hipLaunchKernelGGL(init_ws_kernel, dim3(1), dim3(128), 0, stream, ws);
    hipLaunchKernelGGL(count_kernel, dim3((B + 255) / 256), dim3(256), 0, stream,
                       pos, counts, B);
    hipLaunchKernelGGL(scan_kernel, dim3(1), dim3(1), 0, stream, counts, poffs, E);
    hipLaunchKernelGGL(scatter_kernel, dim3((B + 255) / 256), dim3(256), 0, stream,
                       pos, poffs, cursors, bucket, B);

    const int maxTiles  = (B + 15) / 16 + E;   // padded segments bound
    const int numBlocks = (maxTiles + WAVES_PER_BLOCK - 1) / WAVES_PER_BLOCK;
    hipLaunchKernelGGL(moe_wmma_kernel, dim3(numBlocks), dim3(WAVES_PER_BLOCK * 32),
                       0, stream, x, W1, b1, W0, b0, bucket, counts, poffs,
                       (float*)d_out, E);
}